// _PointwiseSpatialAttention_11192684773452
// MI455X (gfx1250) — compile-verified
//
#include <hip/hip_runtime.h>
#include <math.h>

// ---------------------------------------------------------------------------
// PointwiseSpatialAttention on gfx1250 (MI455X), exact fp32 via
// V_WMMA_F32_16X16X4_F32 with a double-buffered LDS pipeline fed by the
// CDNA5 async-to-LDS engine (GLOBAL_LOAD_ASYNC_TO_LDS_B128 + ASYNCcnt).
//
//   red = relu(bn1(Wred[512,2048] @ X[2048,4096]))
//   a1  = relu(bn2(W1  [512, 512] @ red))
//   A   = softmax_over_rows_per_col( W2[4096,512] @ a1 )
//   fm  = red[512,4096] @ A[4096,4096]      (written straight into d_out)
//
// Roofline: ~0.9 GB of HBM traffic @ 23.3 TB/s dominates the ~180 GFLOP of
// GEMM work, so exact fp32 WMMA costs nothing vs. bf16 and keeps softmax
// numerics bit-faithful.
// ---------------------------------------------------------------------------

typedef __attribute__((ext_vector_type(2))) float v2f;
typedef __attribute__((ext_vector_type(8))) float v8f;
typedef __attribute__((ext_vector_type(4))) int   i4;

#define BN_EPS     1e-5f
#define KB         32            // K-block staged in LDS per iteration
#define BNT        512           // block N tile (8 waves x 64 cols)
#define LDS_STRIDE 520           // 512 + 8 pad: lanes0-15 vs 16-31 rows land
                                 // in disjoint 16-bank groups (2*520 % 64 = 16)

#if defined(__HIP_DEVICE_COMPILE__) && __has_builtin(__builtin_amdgcn_global_load_async_to_lds_b128)
#define USE_ASYNC_LDS 1
// Builtin signature (from hipcc diagnostics): pointer args are int4* in
// addrspace(1) (global source, non-const) and addrspace(3) (LDS dest).
typedef __attribute__((address_space(1))) i4* g1ptr_t;
typedef __attribute__((address_space(3))) i4* l3ptr_t;
#if __has_builtin(__builtin_amdgcn_s_wait_asynccnt)
#define WAIT_ASYNC(n) __builtin_amdgcn_s_wait_asynccnt(n)
#else
#define WAIT_ASYNC(n) asm volatile("s_wait_asynccnt %0" ::"n"(n))
#endif
#else
#define USE_ASYNC_LDS 0
#define WAIT_ASYNC(n) ((void)0)
#endif

// D = W(MxK) * X(KxN), row-major; optional fused BN+ReLU keyed on output row.
// Block: 256 threads = 8 waves; block tile 16 rows x 512 cols; each wave owns
// a 16x64 tile (4 f32 accumulators).  grid = (N/512, M/16, batch).
//
// WMMA f32 16x16x4 fragments (ISA 7.12.2):
//   A 16x4 : lanes 0-15 -> row M=lane,   K={0,1} in v[0],v[1];
//            lanes16-31 -> row M=lane-16,K={2,3}.
//   B 4x16 : lanes 0-15 -> col N=lane,   K={0,1}; lanes16-31 K={2,3}.
//   C/D    : VGPR g: lanes 0-15 -> (row g, col lane); lanes16-31 -> (row g+8,
//            col lane-16).
__global__ __launch_bounds__(256)
void gemm_f32_wmma(const float* __restrict__ W, long long wStride,
                   const float* __restrict__ X, long long xStride,
                   float*       __restrict__ Out, long long oStride,
                   const float* __restrict__ bnp, int Cbn,   // [4,Cbn] or null
                   int M, int K, int N)
{
    __shared__ float Xs[2][KB * LDS_STRIDE];   // 2 x 65 KB double buffer

    const int tid   = threadIdx.x;
    const int wave  = tid >> 5;
    const int lane  = tid & 31;
    const int l16   = lane & 15;
    const int kh    = lane >> 4;                 // 0|1 -> K offset 0|2
    const int tileM = blockIdx.y << 4;
    const int tileN = blockIdx.x * BNT;

    const float* Wb = W   + (size_t)blockIdx.z * (size_t)wStride;
    const float* Xb = X   + (size_t)blockIdx.z * (size_t)xStride;
    float*       Ob = Out + (size_t)blockIdx.z * (size_t)oStride;

    // Cooperative staging geometry: 256 threads move KB x BNT floats as
    // 16-byte chunks; thread covers rows {rt, rt+2, rt+4, ...}.
    const int rt  = tid >> 7;                    // 0..1
    const int c4  = (tid & 127) << 2;            // float col (16B granules)
    const float* gstage = Xb + (size_t)rt * (size_t)N + tileN + c4;
    float*       lstage = &Xs[0][0] + rt * LDS_STRIDE + c4;
    const size_t gStep  = (size_t)2 * (size_t)N; // rows per chunk step

    v8f acc[4] = {v8f{0}, v8f{0}, v8f{0}, v8f{0}};

    const float* wrow = Wb + (size_t)(tileM + l16) * (size_t)K + (kh << 1);

    // ---- stage first K-block into buffer 0 -------------------------------
#if USE_ASYNC_LDS
#pragma unroll
    for (int i = 0; i < 16; ++i)
        __builtin_amdgcn_global_load_async_to_lds_b128(
            (g1ptr_t)(gstage + (size_t)i * gStep),
            (l3ptr_t)(lstage + i * 2 * LDS_STRIDE), 0, 0);
#else
#pragma unroll
    for (int i = 0; i < 16; ++i)
        *(float4*)(lstage + i * 2 * LDS_STRIDE) =
            *(const float4*)(gstage + (size_t)i * gStep);
#endif

    int cur = 0;
    for (int kb = 0; kb < K; kb += KB) {
        // ---- prefetch next K-block into the other buffer -----------------
        if (kb + KB < K) {
            const float* gnxt = gstage + (size_t)(kb + KB) * (size_t)N;
            float*       lnxt = &Xs[cur ^ 1][0] + rt * LDS_STRIDE + c4;
#if USE_ASYNC_LDS
#pragma unroll
            for (int i = 0; i < 16; ++i)
                __builtin_amdgcn_global_load_async_to_lds_b128(
                    (g1ptr_t)(gnxt + (size_t)i * gStep),
                    (l3ptr_t)(lnxt + i * 2 * LDS_STRIDE), 0, 0);
#else
#pragma unroll
            for (int i = 0; i < 16; ++i)
                *(float4*)(lnxt + i * 2 * LDS_STRIDE) =
                    *(const float4*)(gnxt + (size_t)i * gStep);
#endif
            WAIT_ASYNC(16);              // current buffer's 16 copies done
        } else {
            WAIT_ASYNC(0);
        }
        __syncthreads();                 // everyone's current buffer visible

        // ---- preload this block's A fragments (L2-resident weights) ------
        v2f aF[KB / 4];
#pragma unroll
        for (int s = 0; s < KB / 4; ++s)
            aF[s] = *(const v2f*)(wrow + kb + 4 * s);

        // ---- 32 back-to-back WMMAs against LDS ---------------------------
        const float* xs = &Xs[cur][0] + (kh << 1) * LDS_STRIDE + (wave << 6) + l16;
#pragma unroll
        for (int s = 0; s < KB / 4; ++s) {
#pragma unroll
            for (int j = 0; j < 4; ++j) {
                v2f b;
                b.x = xs[(4 * s)     * LDS_STRIDE + j * 16];
                b.y = xs[(4 * s + 1) * LDS_STRIDE + j * 16];
                acc[j] = __builtin_amdgcn_wmma_f32_16x16x4_f32(
                    false, aF[s], false, b, (short)0, acc[j], false, false);
            }
        }
        __syncthreads();                 // done reading before buffer reuse
        cur ^= 1;
    }

    // ---- epilogue: optional BN+ReLU, store 16x64 per wave ----------------
    const int rbase = tileM + (kh << 3);
#pragma unroll
    for (int g = 0; g < 8; ++g) {
        const int row = rbase + g;
        float sc = 1.0f, sh = 0.0f;
        if (bnp) {
            const float gam = bnp[row];
            const float bet = bnp[Cbn + row];
            const float mu  = bnp[2 * Cbn + row];
            const float var = bnp[3 * Cbn + row];
            sc = gam / sqrtf(var + BN_EPS);
            sh = bet - mu * sc;
        }
#pragma unroll
        for (int j = 0; j < 4; ++j) {
            float v = acc[j][g];
            if (bnp) v = fmaxf(v * sc + sh, 0.0f);
            Ob[(size_t)row * (size_t)N + tileN + (wave << 6) + j * 16 + l16] = v;
        }
    }
}

// Column-wise softmax over the C (row) dimension of A[C,N], per batch.
// Online max+sum pass, then normalize; 256 consecutive columns per block.
__global__ __launch_bounds__(256)
void softmax_col(float* __restrict__ A, int C, int N)
{
    float* Ab = A + (size_t)blockIdx.y * (size_t)C * (size_t)N;
    const int t = blockIdx.x * 256 + threadIdx.x;

    float m = -INFINITY, s = 0.0f;
    for (int k = 0; k < C; ++k) {
        const float v  = Ab[(size_t)k * (size_t)N + t];
        const float nm = fmaxf(m, v);
        s = s * __expf(m - nm) + __expf(v - nm);
        m = nm;
    }
    const float inv = 1.0f / s;
    for (int k = 0; k < C; ++k) {
        const size_t idx = (size_t)k * (size_t)N + t;
        Ab[idx] = __expf(Ab[idx] - m) * inv;
    }
}

extern "C" void kernel_launch(void* const* d_in, const int* in_sizes, int n_in,
                              void* d_out, int out_size, void* d_ws, size_t ws_size,
                              hipStream_t stream)
{
    (void)in_sizes; (void)n_in; (void)out_size; (void)ws_size;

    const float* x = (const float*)d_in[0];

    const int NB  = 2;      // batch
    const int CIN = 2048;   // input channels
    const int RC  = 512;    // reduced channels
    const int HW  = 4096;   // 64*64 spatial
    const int CA  = 4096;   // attention channels (== HW)

    // Workspace (floats): red[2,512,4096]  a1[2,512,4096]  A[2,4096,4096]
    float* red = (float*)d_ws;
    float* a1  = red + (size_t)NB * RC * HW;
    float* Am  = a1  + (size_t)NB * RC * HW;

    const dim3 blk(256);

    for (int br = 0; br < 2; ++br) {
        const float* w_red = (const float*)d_in[1 + 5 * br];
        const float* bn1   = (const float*)d_in[2 + 5 * br];
        const float* w1    = (const float*)d_in[3 + 5 * br];
        const float* bn2   = (const float*)d_in[4 + 5 * br];
        const float* w2    = (const float*)d_in[5 + 5 * br];

        // red = relu(bn1(Wred @ X))        M=512 K=2048 N=4096
        gemm_f32_wmma<<<dim3(HW / BNT, RC / 16, NB), blk, 0, stream>>>(
            w_red, 0, x, (long long)CIN * HW, red, (long long)RC * HW,
            bn1, RC, RC, CIN, HW);

        // a1 = relu(bn2(W1 @ red))         M=512 K=512 N=4096
        gemm_f32_wmma<<<dim3(HW / BNT, RC / 16, NB), blk, 0, stream>>>(
            w1, 0, red, (long long)RC * HW, a1, (long long)RC * HW,
            bn2, RC, RC, RC, HW);

        // A = W2 @ a1                      M=4096 K=512 N=4096
        gemm_f32_wmma<<<dim3(HW / BNT, CA / 16, NB), blk, 0, stream>>>(
            w2, 0, a1, (long long)RC * HW, Am, (long long)CA * HW,
            nullptr, 0, CA, RC, HW);

        // softmax over channel (row) dim, per spatial column
        softmax_col<<<dim3(HW / 256, NB), blk, 0, stream>>>(Am, CA, HW);

        // fm = red @ A  -> d_out[:, br*512:(br+1)*512, :, :]
        float* outp = (float*)d_out + (size_t)br * RC * HW;
        gemm_f32_wmma<<<dim3(HW / BNT, RC / 16, NB), blk, 0, stream>>>(
            red, (long long)RC * HW, Am, (long long)CA * HW,
            outp, (long long)2 * RC * HW,
            nullptr, 0, RC, CA, HW);
    }
}